// PaletteViewSynthesis_54966991454709
// MI455X (gfx1250) — compile-verified
//
#include <hip/hip_runtime.h>

typedef float v2f __attribute__((ext_vector_type(2)));
typedef float v8f __attribute__((ext_vector_type(8)));

#define HW        65536             // 256*256
#define BATCH     16
#define NVIEW     8
#define NELEM     (BATCH * 3 * HW)  // mean denominator: 3,145,728
#define NBLOCKS   1024              // 1024 blocks * 8 waves = 8192 waves
#define WAVES_TOT 8192
#define GRP_PER_WAVE 8              // 8192 waves * 8 groups * 16 px = 16*65536 px

// ---------------------------------------------------------------------------
// One wave owns 8 consecutive 16-pixel groups of a single batch image:
//   wid in [0,8192): b = wid>>9 (512 waves per image), pixels wid%512 * 128 ..
//   lanes 0-15  : pixel n (N-column of WMMA B/D)
//   lanes 16-31 : duplicate pixel (carry K=2..3 operands; loads hit the same
//                 cachelines as lanes 0-15, so HBM traffic stays ~ideal)
// einsum 'oc,bvchw->bvohw' = two V_WMMA_F32_16X16X4_F32 per group-view chain:
//   D_base = W[:,4:6] x {ny1,ny2} + (bk + 0.1*mod)      (view-independent)
//   D_v    = W[:,0:4] x {yc0,yc1,yc2,ny0} + D_base      (per view)
// Softmax over views without max-subtraction (logits are O(1); shift-invariant)
// using hardware __expf, accumulated online; MSE reduced wave->block->global.
// ---------------------------------------------------------------------------
__global__ __launch_bounds__(256) void pvs_main_kernel(
    const float* __restrict__ y0,
    const float* __restrict__ ycond,
    const float* __restrict__ angle,
    const float* __restrict__ noise,
    const float* __restrict__ randu,
    const float* __restrict__ Wk,
    const float* __restrict__ bk,
    const float* __restrict__ gammas,
    const int*   __restrict__ viewcnt,
    const int*   __restrict__ tt,
    float* __restrict__ partial)
{
    const int  lane = threadIdx.x & 31;
    const int  p    = lane & 15;
    const bool half = lane >= 16;

    // wave id is identical across the wave -> make it provably uniform
    const int wid = __builtin_amdgcn_readfirstlane(
        (int)((blockIdx.x * blockDim.x + threadIdx.x) >> 5));
    const int b      = wid >> 9;           // batch image (uniform scalar)
    const int hwbase = (wid & 511) << 7;   // first of 128 pixels in this image

    // ---- per-batch scalars (pure SALU / s_load) ----
    const int   tb  = tt[b];
    const float g1  = gammas[tb - 1];
    const float g2  = gammas[tb];
    const float sg  = (g2 - g1) * randu[b] + g1;
    const float sqg = sqrtf(sg);
    const float s1g = sqrtf(1.0f - sg);
    const float addc = 0.1f * (sg + angle[b]);
    const int   vc   = viewcnt[b];

    // ---- hoisted WMMA A operands: Wk rows striped over lanes (rows 4..15=0)
    const bool rowok = (p < 4);
    v2f a1, a2;
    a1.x = rowok ? Wk[p * 6 + (half ? 2 : 0)] : 0.0f;
    a1.y = rowok ? Wk[p * 6 + (half ? 3 : 1)] : 0.0f;
    a2.x = (rowok && !half) ? Wk[p * 6 + 4] : 0.0f;
    a2.y = (rowok && !half) ? Wk[p * 6 + 5] : 0.0f;

    // ---- hoisted C accumulator: bias in rows 0..3 (lanes 0-15 only) ----
    v8f cinit;
    #pragma unroll
    for (int r = 0; r < 8; ++r) cinit[r] = 0.0f;
    if (!half) {
        cinit[0] = bk[0] + addc;
        cinit[1] = bk[1] + addc;
        cinit[2] = bk[2] + addc;
        cinit[3] = bk[3] + addc;
    }

    // ---- per-lane stream base pointers ----
    const float* y0b  = y0    + (size_t)b * (3 * HW) + p;
    const float* nzb  = noise + (size_t)b * (3 * HW) + p;
    const float* ycb0 = ycond + (size_t)b * (NVIEW * 3 * HW) + p
                              + (half ? 2 * HW : 0);          // c0 | c2 plane
    const float* ycb1 = ycond + (size_t)b * (NVIEW * 3 * HW) + p + HW; // c1

    float lsum = 0.0f;

    for (int gi = 0; gi < GRP_PER_WAVE; ++gi) {
        const int hw0 = hwbase + (gi << 4);

        // single-use streams: non-temporal loads
        const float yv0 = __builtin_nontemporal_load(y0b + hw0);
        const float yv1 = __builtin_nontemporal_load(y0b + hw0 + HW);
        const float yv2 = __builtin_nontemporal_load(y0b + hw0 + 2 * HW);
        const float nz0 = __builtin_nontemporal_load(nzb + hw0);
        const float nz1 = __builtin_nontemporal_load(nzb + hw0 + HW);
        const float nz2 = __builtin_nontemporal_load(nzb + hw0 + 2 * HW);
        const float ny0 = sqg * yv0 + s1g * nz0;
        const float ny1 = sqg * yv1 + s1g * nz1;
        const float ny2 = sqg * yv2 + s1g * nz2;

        // view-independent WMMA: D_base = W[:,4:6] x {ny1,ny2} + bias
        v2f b2;
        b2.x = half ? 0.0f : ny1;
        b2.y = half ? 0.0f : ny2;
        const v8f dbase = __builtin_amdgcn_wmma_f32_16x16x4_f32(
            false, a2, false, b2, (short)0, cinit, false, false);

        float ssum = 0.0f, acc0 = 0.0f, acc1 = 0.0f, acc2 = 0.0f;

        #pragma unroll
        for (int v = 0; v < NVIEW; ++v) {
            // B1: V0 = { yc_c0[p] | yc_c2[p] }, V1 = { yc_c1[p] | ny0[p] }
            v2f b1;
            b1.x = __builtin_nontemporal_load(ycb0 + hw0 + v * (3 * HW));
            const float c1v =
                  __builtin_nontemporal_load(ycb1 + hw0 + v * (3 * HW));
            b1.y = half ? ny0 : c1v;

            const v8f d = __builtin_amdgcn_wmma_f32_16x16x4_f32(
                false, a1, false, b1, (short)0, dbase, false, false);

            if (v < vc) {                 // uniform scalar branch
                const float e = __expf(d[3]);   // v_exp_f32; |logit| is O(1)
                ssum += e;
                acc0 += e * d[0];
                acc1 += e * d[1];
                acc2 += e * d[2];
            }
        }

        if (!half) {
            const float inv = 1.0f / ssum;
            const float e0 = nz0 - acc0 * inv;
            const float e1 = nz1 - acc1 * inv;
            const float e2 = nz2 - acc2 * inv;
            lsum += e0 * e0 + e1 * e1 + e2 * e2;
        }
    }

    // ---- reduction: wave shuffle -> LDS -> per-block partial ----
    #pragma unroll
    for (int off = 16; off > 0; off >>= 1)
        lsum += __shfl_down(lsum, off, 32);

    __shared__ float wsums[8];
    if (lane == 0) wsums[threadIdx.x >> 5] = lsum;
    __syncthreads();
    if (threadIdx.x == 0) {
        float s = 0.0f;
        #pragma unroll
        for (int i = 0; i < 8; ++i) s += wsums[i];
        partial[blockIdx.x] = s;
    }
}

// ---------------------------------------------------------------------------
// Final reduction of per-block partials -> mean -> d_out[0]
// ---------------------------------------------------------------------------
__global__ __launch_bounds__(256) void pvs_reduce_kernel(
    const float* __restrict__ partial, int n, float* __restrict__ out)
{
    __shared__ float sh[256];
    float s = 0.0f;
    for (int i = threadIdx.x; i < n; i += 256) s += partial[i];
    sh[threadIdx.x] = s;
    __syncthreads();
    #pragma unroll
    for (int st = 128; st > 0; st >>= 1) {
        if ((int)threadIdx.x < st) sh[threadIdx.x] += sh[threadIdx.x + st];
        __syncthreads();
    }
    if (threadIdx.x == 0) out[0] = sh[0] * (1.0f / (float)NELEM);
}

extern "C" void kernel_launch(void* const* d_in, const int* in_sizes, int n_in,
                              void* d_out, int out_size, void* d_ws, size_t ws_size,
                              hipStream_t stream) {
    const float* y0    = (const float*)d_in[0];
    const float* ycond = (const float*)d_in[1];
    const float* angle = (const float*)d_in[2];
    const float* noise = (const float*)d_in[3];
    const float* randu = (const float*)d_in[4];
    const float* Wk    = (const float*)d_in[5];
    const float* bk    = (const float*)d_in[6];
    const float* gam   = (const float*)d_in[7];
    const int*   vc    = (const int*)d_in[8];
    const int*   t     = (const int*)d_in[9];

    float* partial = (float*)d_ws;   // NBLOCKS floats of scratch

    pvs_main_kernel<<<NBLOCKS, 256, 0, stream>>>(
        y0, ycond, angle, noise, randu, Wk, bk, gam, vc, t, partial);
    pvs_reduce_kernel<<<1, 256, 0, stream>>>(partial, NBLOCKS, (float*)d_out);
}